// pointnetplus_2061584302575
// MI455X (gfx1250) — compile-verified
//
#include <hip/hip_runtime.h>
#include <hip/hip_bf16.h>
#include <float.h>
#include <stdint.h>

typedef __attribute__((ext_vector_type(2))) float v2f;
typedef __attribute__((ext_vector_type(8))) float v8f;
typedef int v4i_b128 __attribute__((vector_size(16)));  // for async-LDS builtin

#if defined(__HIP_DEVICE_COMPILE__) && \
    __has_builtin(__builtin_amdgcn_global_load_async_to_lds_b128)
#define HAVE_ASYNC_LDS 1
#endif

// ---------------------------------------------------------------------------
// CDNA5 f32 WMMA: D(16x16) = A(16x4) * B(4x16) + C   (wave32, K step = 4)
// ---------------------------------------------------------------------------
__device__ inline v8f wmma_f32_4(v2f a, v2f b, v8f c) {
  return __builtin_amdgcn_wmma_f32_16x16x4_f32(
      /*neg_a=*/false, a, /*neg_b=*/false, b,
      /*c_mod=*/(short)0, c, /*reuse_a=*/false, /*reuse_b=*/false);
}

struct LayerP {
  const float* w;   // (O, K) row-major, K padded to multiple of 4
  const float* b;
  const float* g;   // gamma
  const float* be;  // beta
  const float* mn;  // mean
  const float* vr;  // var
};

// ---------------------------------------------------------------------------
// Cooperative global->LDS copy of a contiguous block (bytes % 16 == 0).
// Uses CDNA5 async-to-LDS DMA (ASYNCcnt) when available.
// All 256 threads must participate; caller must __syncthreads() after.
// ---------------------------------------------------------------------------
__device__ inline void cp_lds_async(void* ldsDst, const void* gSrc, int bytes,
                                    int tid) {
#ifdef HAVE_ASYNC_LDS
  const uint32_t lbase = (uint32_t)(uintptr_t)ldsDst;
  const uintptr_t gbase = (uintptr_t)gSrc;
  for (int o = tid * 16; o < bytes; o += 256 * 16) {
    __builtin_amdgcn_global_load_async_to_lds_b128(
        (__attribute__((address_space(1))) v4i_b128*)(gbase + o),
        (__attribute__((address_space(3))) v4i_b128*)(uintptr_t)(lbase + o),
        0, 0);
  }
  asm volatile("s_wait_asynccnt 0x0" ::: "memory");
#else
  float4* d = (float4*)ldsDst;
  const float4* s = (const float4*)gSrc;
  const int n4 = bytes >> 4;
  for (int i = tid; i < n4; i += 256) d[i] = s[i];
#endif
}

// Zero-pad weights (O,Kreal) -> (O,Kpad) so GEMM K-loops need no guards.
__global__ void pad_weights(const float* __restrict__ w, int O, int Kreal,
                            int Kpad, float* __restrict__ out) {
  const int g = blockIdx.x * blockDim.x + threadIdx.x;
  if (g >= O * Kpad) return;
  const int o = g / Kpad;
  const int k = g - o * Kpad;
  out[g] = (k < Kreal) ? w[(long)o * Kreal + k] : 0.0f;
}

// ---------------------------------------------------------------------------
// Farthest point sampling. One block per batch; per-batch point block is
// contiguous for both layouts -> async-stage into LDS, access via strides.
// ---------------------------------------------------------------------------
template <int NMAX>
__global__ __launch_bounds__(256) void fps_kernel(
    const float* __restrict__ pts, long bS, int pS, int cS,
    int N, int S, float* __restrict__ newxyz /* (B,S,3) */) {
  __shared__ __align__(16) float sp[NMAX * 3];
  __shared__ float sd[NMAX];
  __shared__ float rv[256];
  __shared__ int ri[256];
  __shared__ int sFar;

  const int b = blockIdx.x;
  const int tid = threadIdx.x;
  cp_lds_async(sp, pts + (long)b * bS, N * 3 * 4, tid);
  for (int i = tid; i < N; i += 256) sd[i] = 1e10f;
  if (tid == 0) sFar = 0;
  __syncthreads();

  for (int j = 0; j < S; ++j) {
    const int far = sFar;
    if (tid == 0) {
      newxyz[((long)b * S + j) * 3 + 0] = sp[far * pS + 0 * cS];
      newxyz[((long)b * S + j) * 3 + 1] = sp[far * pS + 1 * cS];
      newxyz[((long)b * S + j) * 3 + 2] = sp[far * pS + 2 * cS];
    }
    const float cx = sp[far * pS + 0 * cS];
    const float cy = sp[far * pS + 1 * cS];
    const float cz = sp[far * pS + 2 * cS];
    float best = -1.0f;
    int bi = 0;
    for (int i = tid; i < N; i += 256) {
      const float dx = sp[i * pS + 0 * cS] - cx;
      const float dy = sp[i * pS + 1 * cS] - cy;
      const float dz = sp[i * pS + 2 * cS] - cz;
      const float nd = fminf(sd[i], dx * dx + dy * dy + dz * dz);
      sd[i] = nd;
      if (nd > best) { best = nd; bi = i; }
    }
    rv[tid] = best;
    ri[tid] = bi;
    __syncthreads();
    for (int off = 128; off > 0; off >>= 1) {
      if (tid < off) {
        const float ov = rv[tid + off];
        const int oi = ri[tid + off];
        if (ov > rv[tid] || (ov == rv[tid] && oi < ri[tid])) {
          rv[tid] = ov; ri[tid] = oi;
        }
      }
      __syncthreads();
    }
    if (tid == 0) sFar = ri[0];
    __syncthreads();
  }
}

// ---------------------------------------------------------------------------
// Ball query: one wave32 per query point. Wave ballot + ctz extracts the
// first-nsample in-radius indices in ascending order (== sort-truncate of
// the reference); pads with the first hit.
// ---------------------------------------------------------------------------
template <int NMAX>
__global__ __launch_bounds__(256) void ball_query_kernel(
    const float* __restrict__ pts, long bS, int pS, int cS, int N,
    const float* __restrict__ newxyz, int S, float r2, int nsample,
    int* __restrict__ gidx) {
  __shared__ __align__(16) float sp[NMAX * 3];
  const int tid = threadIdx.x;
  const int wave = tid >> 5;
  const int lane = tid & 31;
  const int blocksPerBatch = S / 8;
  const int b = blockIdx.x / blocksPerBatch;
  const int sBase = (blockIdx.x % blocksPerBatch) * 8;

  cp_lds_async(sp, pts + (long)b * bS, N * 3 * 4, tid);
  __syncthreads();

  const int s = sBase + wave;
  const float qx = newxyz[((long)b * S + s) * 3 + 0];
  const float qy = newxyz[((long)b * S + s) * 3 + 1];
  const float qz = newxyz[((long)b * S + s) * 3 + 2];
  int* og = gidx + ((long)b * S + s) * nsample;

  int count = 0;
  int first = 0;
  for (int baseI = 0; baseI < N && count < nsample; baseI += 32) {
    const int i = baseI + lane;
    bool ok = false;
    if (i < N) {
      const float dx = sp[i * pS + 0 * cS] - qx;
      const float dy = sp[i * pS + 1 * cS] - qy;
      const float dz = sp[i * pS + 2 * cS] - qz;
      ok = (dx * dx + dy * dy + dz * dz) <= r2;
    }
    unsigned int mask = __builtin_amdgcn_ballot_w32(ok);
    while (mask != 0u && count < nsample) {
      const int bit = __builtin_ctz(mask);
      mask &= mask - 1u;
      const int idx = baseI + bit;
      if (count == 0) first = idx;
      if (lane == 0) og[count] = idx;
      ++count;
    }
  }
  if (lane == 0) {
    for (int j = count; j < nsample; ++j) og[j] = first;
  }
}

// ---------------------------------------------------------------------------
// One MLP layer, LDS -> LDS, via f32 WMMA. out = relu(bn(in @ W^T + b)).
// K compile-time, multiple of 4, weights pre-padded. Each wave owns one
// 16-row M-tile, preloads its whole A K-strip into registers, then streams
// B over the N-tiles: 1 global b64 load per WMMA.
// ---------------------------------------------------------------------------
template <int K>
__device__ inline void lds_mlp_layer(
    const float* ldsIn, int inStride,
    const LayerP lp, int N, int rows,
    float* ldsOut, int outStride, int tid) {
  const int wave = tid >> 5;
  const int lane = tid & 31;
  const int ln = lane & 15;
  const int kh = lane >> 4;
  const int mT = rows >> 4;       // 8 or 4 (power of two)
  const int nT = N >> 4;
  const int wpm = 8 / mT;         // waves per m-tile
  const int m0 = (wave % mT) << 4;
  const int nPer = nT / wpm;
  const int nStart = (wave / mT) * nPer;

  float aF[K / 2];
  const float* aRow = ldsIn + (m0 + ln) * inStride + (kh << 1);
#pragma unroll
  for (int ks = 0; ks < K / 4; ++ks) {
    aF[2 * ks] = aRow[4 * ks];
    aF[2 * ks + 1] = aRow[4 * ks + 1];
  }

  for (int nt = nStart; nt < nStart + nPer; ++nt) {
    const int n0 = nt << 4;
    v8f acc = {};
    const float* wRow = lp.w + (long)(n0 + ln) * K + (kh << 1);
#pragma unroll 4
    for (int ks = 0; ks < K / 4; ++ks) {
      v2f a;
      a.x = aF[2 * ks];
      a.y = aF[2 * ks + 1];
      const v2f bb = *(const v2f*)(wRow + 4 * ks);  // global_load_b64
      acc = wmma_f32_4(a, bb, acc);
    }
    const int o = n0 + ln;
    const float sc = lp.g[o] * rsqrtf(lp.vr[o] + 1e-5f);
    const float sh = (lp.b[o] - lp.mn[o]) * sc + lp.be[o];
#pragma unroll
    for (int r = 0; r < 8; ++r) {
      const int mm = m0 + r + (kh << 3);
      ldsOut[mm * outStride + o] = fmaxf(acc[r] * sc + sh, 0.0f);
    }
  }
}

// ---------------------------------------------------------------------------
// Fused set-abstraction: gather(+center,+concat feats) -> 3x(WMMA+BN+ReLU)
// entirely in LDS -> maxpool over nsample -> pooled (B,S,C3).
// ---------------------------------------------------------------------------
template <int TILE_S, int NS, int CINP, int FEATC, int C1, int C2, int C3>
__global__ __launch_bounds__(256) void sa_fused_kernel(
    const float* __restrict__ src, long bS, long pS, long cS, int srcN,
    const float* __restrict__ srcFeat,            // (B, srcN, FEATC) or null
    const float* __restrict__ newxyz, int S,      // (B,S,3)
    const int* __restrict__ gidx,                 // (B,S,NS)
    LayerP L1, LayerP L2, LayerP L3,
    float* __restrict__ pooled)                   // (B,S,C3)
{
  constexpr int ROWS = TILE_S * NS;
  constexpr int MAXA = (CINP > C2) ? CINP : C2;
  constexpr int MAXB = (C1 > C3) ? C1 : C3;
  constexpr int SA = MAXA + 5;  // odd strides -> no LDS bank conflicts
  constexpr int SB = MAXB + 5;
  __shared__ float bufA[ROWS * SA];
  __shared__ float bufB[ROWS * SB];
  __shared__ int sIdx[ROWS];

  const int tid = threadIdx.x;
  const int blocksPerBatch = S / TILE_S;
  const int b = blockIdx.x / blocksPerBatch;
  const int s0 = (blockIdx.x % blocksPerBatch) * TILE_S;

  for (int r = tid; r < ROWS; r += 256) {
    const int sL = r / NS;
    const int k = r - sL * NS;
    sIdx[r] = gidx[((long)b * S + s0 + sL) * NS + k];
  }
  __syncthreads();

  // Gather grouped, centered input (+ features) into bufA (zero-padded K).
  for (int e = tid; e < ROWS * CINP; e += 256) {
    const int r = e / CINP;
    const int c = e - r * CINP;
    const int sL = r / NS;
    const int s = s0 + sL;
    const int idx = sIdx[r];
    float v = 0.0f;
    if (c < 3) {
      v = src[(long)b * bS + (long)idx * pS + (long)c * cS] -
          newxyz[((long)b * S + s) * 3 + c];
    } else if (FEATC > 0 && c < 3 + FEATC) {
      v = srcFeat[((long)b * srcN + idx) * FEATC + (c - 3)];
    }
    bufA[r * SA + c] = v;
  }
  __syncthreads();

  lds_mlp_layer<CINP>(bufA, SA, L1, C1, ROWS, bufB, SB, tid);
  __syncthreads();
  lds_mlp_layer<C1>(bufB, SB, L2, C2, ROWS, bufA, SA, tid);
  __syncthreads();
  lds_mlp_layer<C2>(bufA, SA, L3, C3, ROWS, bufB, SB, tid);
  __syncthreads();

  // Max over nsample.
  for (int e = tid; e < TILE_S * C3; e += 256) {
    const int sL = e / C3;
    const int c = e - sL * C3;
    const float* col = bufB + (sL * NS) * SB + c;
    float m = col[0];
    for (int j = 1; j < NS; ++j) m = fmaxf(m, col[j * SB]);
    pooled[((long)b * S + s0 + sL) * C3 + c] = m;
  }
}

// ---------------------------------------------------------------------------
// Generic WMMA GEMM: out(M,N) = relu(bn(X(M,K) @ W(N,K)^T + b)).
// K multiple of 4, W pre-padded; A-rows clamped (no predicated loads).
// Each wave computes two N-adjacent 16x16 tiles sharing one A fragment:
// per K-step = 1 A load + 2 B loads + 2 WMMA.
// ---------------------------------------------------------------------------
__global__ __launch_bounds__(256) void gemm_bn_relu(
    const float* __restrict__ X, int M, int K,
    LayerP lp, int N, float* __restrict__ out) {
  const int tid = threadIdx.x;
  const int wave = tid >> 5;
  const int lane = tid & 31;
  const int ln = lane & 15;
  const int kh = lane >> 4;
  const int m0 = blockIdx.x * 64 + (wave & 3) * 16;
  const int n0a = blockIdx.y * 64 + (wave >> 2) * 32;
  const int n0b = n0a + 16;

  v8f acc0 = {}, acc1 = {};
  const int arow = min(m0 + ln, M - 1);  // clamp: loads always valid
  const float* xr = X + (long)arow * K + (kh << 1);
  const float* w0 = lp.w + (long)(n0a + ln) * K + (kh << 1);
  const float* w1 = lp.w + (long)(n0b + ln) * K + (kh << 1);
#pragma unroll 4
  for (int k = 0; k < K; k += 4) {
    const v2f a = *(const v2f*)(xr + k);
    const v2f b0 = *(const v2f*)(w0 + k);
    const v2f b1 = *(const v2f*)(w1 + k);
    acc0 = wmma_f32_4(a, b0, acc0);
    acc1 = wmma_f32_4(a, b1, acc1);
  }

  const int oa = n0a + ln;
  const int ob = n0b + ln;
  const float sca = lp.g[oa] * rsqrtf(lp.vr[oa] + 1e-5f);
  const float sha = (lp.b[oa] - lp.mn[oa]) * sca + lp.be[oa];
  const float scb = lp.g[ob] * rsqrtf(lp.vr[ob] + 1e-5f);
  const float shb = (lp.b[ob] - lp.mn[ob]) * scb + lp.be[ob];
#pragma unroll
  for (int r = 0; r < 8; ++r) {
    const int mm = m0 + r + (kh << 3);
    if (mm < M) {
      out[(long)mm * N + oa] = fmaxf(acc0[r] * sca + sha, 0.0f);
      out[(long)mm * N + ob] = fmaxf(acc1[r] * scb + shb, 0.0f);
    }
  }
}

// ---------------------------------------------------------------------------
// Build SA3 (group_all) input: X3[(b*128+p), 0:260] = [l2_xyz(3), l2(256), 0]
// ---------------------------------------------------------------------------
__global__ void build_x3(const float* __restrict__ l2xyz,
                         const float* __restrict__ l2feat,
                         float* __restrict__ X3, int total) {
  const int g = blockIdx.x * blockDim.x + threadIdx.x;
  if (g >= total) return;
  const int c = g % 260;
  const int p = g / 260;
  float v = 0.0f;
  if (c < 3) v = l2xyz[(long)p * 3 + c];
  else if (c < 259) v = l2feat[(long)p * 256 + (c - 3)];
  X3[g] = v;
}

// out[b*C + c] = max_j X[(b*rowsPerB + j)*C + c]
__global__ void maxpool_rows(const float* __restrict__ X, int rowsPerB, int C,
                             float* __restrict__ out) {
  const int g = blockIdx.x * blockDim.x + threadIdx.x;
  const int b = g / C;
  const int c = g % C;
  const float* p = X + (long)b * rowsPerB * C + c;
  float m = p[0];
  for (int j = 1; j < rowsPerB; ++j) m = fmaxf(m, p[(long)j * C]);
  out[(long)b * C + c] = m;
}

// ---------------------------------------------------------------------------
extern "C" void kernel_launch(void* const* d_in, const int* in_sizes, int n_in,
                              void* d_out, int out_size, void* d_ws,
                              size_t ws_size, hipStream_t stream) {
  (void)in_sizes; (void)n_in; (void)out_size; (void)ws_size;
  const int B = 32, N = 4096;
  const float* xyz = (const float*)d_in[0];  // (B,3,N) planar

  auto L = [&](int base) {
    LayerP p;
    p.w  = (const float*)d_in[base + 0];
    p.b  = (const float*)d_in[base + 1];
    p.g  = (const float*)d_in[base + 2];
    p.be = (const float*)d_in[base + 3];
    p.mn = (const float*)d_in[base + 4];
    p.vr = (const float*)d_in[base + 5];
    return p;
  };
  LayerP sa1l0 = L(1),  sa1l1 = L(7),  sa1l2 = L(13);
  LayerP sa2l0 = L(19), sa2l1 = L(25), sa2l2 = L(31);
  LayerP sa3l0 = L(37), sa3l1 = L(43), sa3l2 = L(49);
  LayerP fc1 = {(const float*)d_in[55], (const float*)d_in[56],
                (const float*)d_in[59], (const float*)d_in[60],
                (const float*)d_in[61], (const float*)d_in[62]};
  LayerP fc2 = {(const float*)d_in[57], (const float*)d_in[58],
                (const float*)d_in[63], (const float*)d_in[64],
                (const float*)d_in[65], (const float*)d_in[66]};

  char* ws = (char*)d_ws;
  size_t off = 0;
  auto carve = [&](size_t bytes) {
    void* p = ws + off;
    off += (bytes + 255) & ~(size_t)255;
    return p;
  };
  float* newxyz1 = (float*)carve((size_t)B * 512 * 3 * 4);
  int*   gidx1   = (int*)  carve((size_t)B * 512 * 32 * 4);
  float* l1feat  = (float*)carve((size_t)B * 512 * 128 * 4);
  float* newxyz2 = (float*)carve((size_t)B * 128 * 3 * 4);
  int*   gidx2   = (int*)  carve((size_t)B * 128 * 64 * 4);
  float* l2feat  = (float*)carve((size_t)B * 128 * 256 * 4);
  float* X3      = (float*)carve((size_t)B * 128 * 260 * 4);
  float* A1      = (float*)carve((size_t)B * 128 * 256 * 4);
  float* A2      = (float*)carve((size_t)B * 128 * 512 * 4);
  float* A3      = (float*)carve((size_t)B * 128 * 1024 * 4);
  float* x1      = (float*)carve((size_t)B * 512 * 4);
  float* w1p     = (float*)carve((size_t)64 * 4 * 4);     // sa1.L0 padded
  float* w2p     = (float*)carve((size_t)128 * 132 * 4);  // sa2.L0 padded
  float* w3p     = (float*)carve((size_t)256 * 260 * 4);  // sa3.L0 padded

  // ---- Pre-pad ragged-K weights so all GEMM K-loops are unconditional.
  pad_weights<<<(64 * 4 + 255) / 256, 256, 0, stream>>>(sa1l0.w, 64, 3, 4, w1p);
  pad_weights<<<(128 * 132 + 255) / 256, 256, 0, stream>>>(sa2l0.w, 128, 131,
                                                           132, w2p);
  pad_weights<<<(256 * 260 + 255) / 256, 256, 0, stream>>>(sa3l0.w, 256, 259,
                                                           260, w3p);
  LayerP sa1l0p = sa1l0; sa1l0p.w = w1p;
  LayerP sa2l0p = sa2l0; sa2l0p.w = w2p;
  LayerP sa3l0p = sa3l0; sa3l0p.w = w3p;

  // ---- Stage 1: xyz (B,3,4096) -> l1_xyz (B,512,3), l1 (B,512,128)
  fps_kernel<4096><<<B, 256, 0, stream>>>(xyz, (long)3 * N, 1, N, N, 512,
                                          newxyz1);
  ball_query_kernel<4096><<<B * (512 / 8), 256, 0, stream>>>(
      xyz, (long)3 * N, 1, N, N, newxyz1, 512, 0.2f * 0.2f, 32, gidx1);
  sa_fused_kernel<4, 32, 4, 0, 64, 64, 128><<<B * (512 / 4), 256, 0, stream>>>(
      xyz, (long)3 * N, 1L, (long)N, N, nullptr, newxyz1, 512, gidx1,
      sa1l0p, sa1l1, sa1l2, l1feat);

  // ---- Stage 2: l1 -> l2_xyz (B,128,3), l2 (B,128,256)
  fps_kernel<512><<<B, 256, 0, stream>>>(newxyz1, (long)512 * 3, 3, 1, 512,
                                         128, newxyz2);
  ball_query_kernel<512><<<B * (128 / 8), 256, 0, stream>>>(
      newxyz1, (long)512 * 3, 3, 1, 512, newxyz2, 128, 0.4f * 0.4f, 64,
      gidx2);
  sa_fused_kernel<1, 64, 132, 128, 128, 128, 256>
      <<<B * 128, 256, 0, stream>>>(newxyz1, (long)512 * 3, 3L, 1L, 512,
                                    l1feat, newxyz2, 128, gidx2,
                                    sa2l0p, sa2l1, sa2l2, l2feat);

  // ---- Stage 3: group_all MLP [256,512,1024], maxpool over 128 pts
  build_x3<<<(B * 128 * 260 + 255) / 256, 256, 0, stream>>>(newxyz2, l2feat,
                                                            X3, B * 128 * 260);
  gemm_bn_relu<<<dim3(4096 / 64, 256 / 64), 256, 0, stream>>>(X3, 4096, 260,
                                                              sa3l0p, 256, A1);
  gemm_bn_relu<<<dim3(4096 / 64, 512 / 64), 256, 0, stream>>>(A1, 4096, 256,
                                                              sa3l1, 512, A2);
  gemm_bn_relu<<<dim3(4096 / 64, 1024 / 64), 256, 0, stream>>>(A2, 4096, 512,
                                                               sa3l2, 1024, A3);

  float* feat = (float*)d_out + (size_t)B * 256;  // second output, (B,1024)
  maxpool_rows<<<(B * 1024) / 256, 256, 0, stream>>>(A3, 128, 1024, feat);

  // ---- FC head: x = relu(bn2(relu(bn1(feat@W1^T+b1))@W2^T+b2))
  gemm_bn_relu<<<dim3(1, 512 / 64), 256, 0, stream>>>(feat, B, 1024, fc1, 512,
                                                      x1);
  gemm_bn_relu<<<dim3(1, 256 / 64), 256, 0, stream>>>(x1, B, 512, fc2, 256,
                                                      (float*)d_out);
}